// Composer_81707457839192
// MI455X (gfx1250) — compile-verified
//
#include <hip/hip_runtime.h>
#include <hip/hip_bf16.h>
#include <stdint.h>

#define N_BATCH 131072
#define N_MODES 32
#define E_CONST 2.71828182845904523536f

#define S1_BLOCKS 64
#define S1_THREADS 256
#define S1_STRIDE (S1_BLOCKS * S1_THREADS)          // 16384, multiple of 32

#define MAIN_BLOCKS 2048
#define MAIN_THREADS 256
#define MAIN_NWAVES ((MAIN_BLOCKS * MAIN_THREADS) / 32)   // 16384 -> 8 rows per wave

// ---------------- Stage 1: per-block partial min over batch, per mode ----------------
__global__ __launch_bounds__(S1_THREADS) void min_stage1(
        const float* __restrict__ phi, float* __restrict__ partial) {
    __shared__ float sm[S1_THREADS];
    const int tid = blockIdx.x * S1_THREADS + threadIdx.x;
    float v = 3.402823466e38f;
    // stride is a compile-time constant: trip count 256, no runtime division
    for (int e = tid; e < N_BATCH * N_MODES; e += S1_STRIDE) v = fminf(v, phi[e]);
    sm[threadIdx.x] = v;
    __syncthreads();
    if (threadIdx.x < 32) {                          // thread t reduces mode t across 8 rows
        float m = sm[threadIdx.x];
        #pragma unroll
        for (int r = 1; r < 8; ++r) m = fminf(m, sm[r * 32 + threadIdx.x]);
        partial[blockIdx.x * 32 + threadIdx.x] = m;
    }
}

// ---------------- Stage 2: fold 64 partials -> mode_min[32] ----------------
__global__ __launch_bounds__(32) void min_stage2(
        const float* __restrict__ partial, float* __restrict__ mode_min) {
    float m = 3.402823466e38f;
    #pragma unroll 8
    for (int r = 0; r < S1_BLOCKS; ++r) m = fminf(m, partial[r * 32 + threadIdx.x]);
    mode_min[threadIdx.x] = m;
}

// ---------------- Main: one wave32 per batch row, lane = mode ----------------
__global__ __launch_bounds__(MAIN_THREADS) void modal_alphas_kernel(
        const float* __restrict__ phi, const float* __restrict__ Gamma,
        const float* __restrict__ w, const float* __restrict__ mode_min,
        float* __restrict__ out) {
    __shared__ float sGam[N_MODES * N_MODES];

    // CDNA5 async global->LDS DMA: 256 lanes x 16B = 4KB (the whole Gamma matrix)
    {
        const float* src = Gamma + threadIdx.x * 4;
        uint32_t lds = (uint32_t)(uintptr_t)(sGam + threadIdx.x * 4);
        asm volatile("global_load_async_to_lds_b128 %0, %1, off"
                     :: "v"(lds), "v"(src) : "memory");
        asm volatile("s_wait_asynccnt 0" ::: "memory");
    }
    __syncthreads();

    const int lane = threadIdx.x & 31;
    const int wid  = (blockIdx.x * MAIN_THREADS + threadIdx.x) >> 5;

    // Lane i keeps row i of the symmetrized Gamma in registers (hoisted out of n-loop)
    float g[N_MODES];
    #pragma unroll
    for (int j = 0; j < N_MODES; ++j) {
        const int r = (j < lane) ? j : lane;
        const int c = (j < lane) ? lane : j;
        g[j] = sGam[r * N_MODES + c];
    }
    const float mmin = mode_min[lane];
    const float wi   = w[lane];
    const float negt = -0.17677669529663687f;        // -1/sqrt(32)

    // compile-time stride: exactly 8 iterations per wave
    for (int n = wid; n < N_BATCH; n += MAIN_NWAVES) {
        float p = phi[n * N_MODES + lane] - mmin + E_CONST;
        p = fmaxf(p, E_CONST);
        const float L = __log2f(p);                  // log2(phic_i), one v_log_f32

        // E_i = sum_j 2^( L_j + g_ij*(L_i - L_j) ); diagonal term is exactly 2^L_i
        float acc = 0.0f;
        #pragma unroll
        for (int j = 0; j < N_MODES; ++j) {
            const float Lj = __uint_as_float(
                __builtin_amdgcn_readlane(__float_as_uint(L), j)); // SGPR broadcast
            acc += __builtin_amdgcn_exp2f(fmaf(g[j], L - Lj, Lj)); // v_fma + v_exp_f32
        }
        const float Ei = acc - __builtin_amdgcn_exp2f(L) + (1.0f + wi) * p;

        // log-softmax across the wave (lanes = modes)
        const float s = negt * Ei;
        float m = s;
        #pragma unroll
        for (int off = 16; off; off >>= 1) m = fmaxf(m, __shfl_xor(m, off, 32));
        const float sm = s - m;
        float e = __expf(sm);
        float Z = e;
        #pragma unroll
        for (int off = 16; off; off >>= 1) Z += __shfl_xor(Z, off, 32);
        const float logit = sm - __logf(Z);

        out[n * N_MODES + lane] = __expf(logit);                    // alphas
        out[N_BATCH * N_MODES + n * N_MODES + lane] = logit;        // logits
    }
}

extern "C" void kernel_launch(void* const* d_in, const int* in_sizes, int n_in,
                              void* d_out, int out_size, void* d_ws, size_t ws_size,
                              hipStream_t stream) {
    const float* phi   = (const float*)d_in[0];   // [131072, 32]
    const float* Gamma = (const float*)d_in[1];   // [32, 32]
    const float* w     = (const float*)d_in[2];   // [32]
    float* out = (float*)d_out;                   // [2 * 131072 * 32]
    float* wsf = (float*)d_ws;
    float* mode_min = wsf;                        // [32]
    float* partial  = wsf + 32;                   // [64 * 32]

    min_stage1<<<S1_BLOCKS, S1_THREADS, 0, stream>>>(phi, partial);
    min_stage2<<<1, 32, 0, stream>>>(partial, mode_min);
    modal_alphas_kernel<<<MAIN_BLOCKS, MAIN_THREADS, 0, stream>>>(phi, Gamma, w, mode_min, out);
}